// RddTransformer_18442589569744
// MI455X (gfx1250) — compile-verified
//
#include <hip/hip_runtime.h>

typedef float v2f __attribute__((ext_vector_type(2)));
typedef float v8f __attribute__((ext_vector_type(8)));

#define BB      8
#define NN      8192
#define DD      768
#define CC      8
#define KK      2
#define CHUNKS  48      // DD / 16
#define NSEG    16      // N segments (waves along N)
#define SEGLEN  512     // NN / NSEG
#define KSTEPS  128     // SEGLEN / 4  (K=4 per WMMA)

// ---------------------------------------------------------------------------
// Kernel 1: per-(batch, d-chunk, n-segment) partial cluster sums via
// V_WMMA_F32_16X16X4_F32:  acc[16 clusters x 16 features] += A(onehot) x B(X)
// ---------------------------------------------------------------------------
__global__ __launch_bounds__(256)
void seg_wmma_kernel(const float* __restrict__ X, const int* __restrict__ idcs,
                     float* __restrict__ part) {
    const int blk   = blockIdx.x;
    const int b     = blk / (CHUNKS * (NSEG / 8));
    const int rem   = blk % (CHUNKS * (NSEG / 8));
    const int chunk = rem / (NSEG / 8);
    const int sgrp  = rem % (NSEG / 8);
    const int wave  = threadIdx.x >> 5;
    const int lane  = threadIdx.x & 31;
    const int seg   = sgrp * 8 + wave;

    const int n0 = seg * SEGLEN;
    const int d0 = chunk * 16;
    const float* Xp = X + ((size_t)b * NN + n0) * DD + d0;
    const int*   ip = idcs + b * NN + n0;

    const int row = lane & 15;   // A: M row / B: N column
    const int hi  = lane >> 4;   // selects K pair {0,1} vs {2,3}

    v8f acc = {};
    for (int s = 0; s < KSTEPS; ++s) {
        const int n = s * 4;
        // 4 cluster indices for this K-step: uniform address -> scalar load
        const int4 ci = *(const int4*)(ip + n);
        const int sel0 = hi ? ci.z : ci.x;   // K = 2*hi + 0
        const int sel1 = hi ? ci.w : ci.y;   // K = 2*hi + 1

        // A-tile (16x4 one-hot, 2 VGPRs): A[row, k] = (cluster[k] == row)
        v2f a;
        a.x = (sel0 == row) ? 1.0f : 0.0f;
        a.y = (sel1 == row) ? 1.0f : 0.0f;

        // B-tile (4x16 of inst_feat, 2 VGPRs): B[k, col] = X[n0+n+k, d0+col]
        const float* rp = Xp + (size_t)(n + 2 * hi) * DD + row;
        v2f bm;
        bm.x = rp[0];    // K = 2*hi
        bm.y = rp[DD];   // K = 2*hi + 1

        acc = __builtin_amdgcn_wmma_f32_16x16x4_f32(
            /*neg_a=*/false, a, /*neg_b=*/false, bm,
            /*c_mod=*/(short)0, acc, /*reuse_a=*/false, /*reuse_b=*/false);
    }

    // Partial tile: part[((b*CHUNKS+chunk)*NSEG+seg)*256 + lane*8 + r]
    //   == sums_partial[cluster = r + 8*hi][d = d0 + (lane&15)]
    float* outp = part + ((((size_t)b * CHUNKS + chunk) * NSEG + seg) * 256) + lane * 8;
    *(v8f*)outp = acc;
}

// ---------------------------------------------------------------------------
// Kernel 2: per-batch counts, seg-reduction, mean, head GEMM, softmax/argmax
// ---------------------------------------------------------------------------
__global__ __launch_bounds__(256)
void finalize_kernel(const int* __restrict__ idcs, const float* __restrict__ W,
                     const float* __restrict__ bias, const float* __restrict__ part,
                     float* __restrict__ out) {
    const int b = blockIdx.x;
    const int t = threadIdx.x;

    __shared__ int   s_cnt[CC];
    __shared__ float s_logit[CC * KK];
    if (t < CC)      s_cnt[t]   = 0;
    if (t < CC * KK) s_logit[t] = bias[t & (KK - 1)];
    __syncthreads();

    // cluster counts
    int cnt[CC];
    for (int c = 0; c < CC; ++c) cnt[c] = 0;
    for (int n = t; n < NN; n += 256) cnt[idcs[b * NN + n]]++;
    for (int c = 0; c < CC; ++c)
        if (cnt[c]) atomicAdd(&s_cnt[c], cnt[c]);
    __syncthreads();

    float invc[CC];
    for (int c = 0; c < CC; ++c)
        invc[c] = 1.0f / fmaxf((float)s_cnt[c], 1.0f);

    // reduce partials over segments, form means, multiply by W (per-thread partials)
    float pl[CC * KK];
    for (int i = 0; i < CC * KK; ++i) pl[i] = 0.0f;
    for (int d = t; d < DD; d += 256) {
        const float w0 = W[d];
        const float w1 = W[DD + d];
        const float* pbase = part + (((size_t)b * CHUNKS + (d >> 4)) * NSEG) * 256
                                  + (d & 15) * 8;
        for (int c = 0; c < CC; ++c) {
            float s = 0.0f;
            for (int g = 0; g < NSEG; ++g) s += pbase[g * 256 + c];
            const float m = s * invc[c];
            pl[c * 2 + 0] += m * w0;
            pl[c * 2 + 1] += m * w1;
        }
    }
    for (int i = 0; i < CC * KK; ++i) atomicAdd(&s_logit[i], pl[i]);
    __syncthreads();

    if (t == 0) {
        int   best = 0;
        float bestScore = -1e30f;
        for (int c = 0; c < CC; ++c) {
            const float l0 = s_logit[2 * c + 0];
            const float l1 = s_logit[2 * c + 1];
            const float mx = fmaxf(l0, l1);
            const float e0 = __expf(l0 - mx);
            const float e1 = __expf(l1 - mx);
            const float score = 1.0f - e0 / (e0 + e1);
            if (score > bestScore) { bestScore = score; best = c; }  // first max wins
        }
        out[b * 2 + 0] = s_logit[2 * best + 0];
        out[b * 2 + 1] = s_logit[2 * best + 1];
    }
}

// ---------------------------------------------------------------------------
extern "C" void kernel_launch(void* const* d_in, const int* in_sizes, int n_in,
                              void* d_out, int out_size, void* d_ws, size_t ws_size,
                              hipStream_t stream) {
    const float* X    = (const float*)d_in[0];   // inst_feat [B,N,D] f32
    const float* W    = (const float*)d_in[1];   // W [K,D] f32
    const float* bias = (const float*)d_in[2];   // b [K] f32
    const int*   idcs = (const int*)d_in[3];     // clusters_idcs [B,N] i32
    float* out  = (float*)d_out;                 // [B,K] f32
    float* part = (float*)d_ws;                  // needs 8*48*16*256*4 = 6 MB

    seg_wmma_kernel<<<dim3(BB * CHUNKS * (NSEG / 8)), 256, 0, stream>>>(X, idcs, part);
    finalize_kernel<<<dim3(BB), 256, 0, stream>>>(idcs, W, bias, part, out);
}